// BinaryCQV_End2End_34797825032439
// MI455X (gfx1250) — compile-verified
//
#include <hip/hip_runtime.h>
#include <math.h>

// ---------------------------------------------------------------------------
// 10-qubit batched state-vector simulator for MI455X (gfx1250).
// State per sample = 32x32 complex f32 matrix in LDS (high qubits = rows,
// low qubits = cols). Rotations become 32x32 complex GEMMs executed with
// V_WMMA_F32_16X16X4_F32; the CNOT chain is a prefix-XOR basis permutation.
// One wave (32 threads) per sample, 2048 single-wave workgroups.
// ---------------------------------------------------------------------------

#define NQ    10
#define DIM   49
#define NH    32     // 2^5, matrix edge
#define ST    33     // padded LDS row stride (floats) to avoid bank conflicts
#define PI_F  3.14159265358979323846f

typedef __attribute__((ext_vector_type(2))) float v2f;
typedef __attribute__((ext_vector_type(8))) float v8f;

__device__ __forceinline__ v8f wmma4(v2f a, v2f b, v8f c) {
  // 8-arg form: (neg_a, A, neg_b, B, c_mod, C, reuse_a, reuse_b)
  // F32 WMMA supports only C-negate in NEG; pass false for A/B.
  return __builtin_amdgcn_wmma_f32_16x16x4_f32(false, a, false, b,
                                               (short)0, c, false, false);
}

// D = A * B, 32x32 complex, all operands/results in LDS planes (row stride ST).
// Fragment layouts per CDNA5 ISA 7.12.2 (wave32):
//   A 16x4 : lane&15 = M, lane>>4 selects K pair {0,1}|{2,3}, regs = within pair
//   B 4x16 : lane&15 = N, lane>>4 selects K pair, regs = within pair
//   C/D    : reg r = row r + 8*(lane>>4), lane&15 = col
__device__ __forceinline__ void cgemm32(const float* __restrict__ Are,
                                        const float* __restrict__ Aim,
                                        const float* __restrict__ Bre,
                                        const float* __restrict__ Bim,
                                        float* __restrict__ Dre,
                                        float* __restrict__ Dim,
                                        int lane) {
  const int m0   = lane & 15;
  const int hsel = lane >> 4;       // lane half
  #pragma unroll
  for (int ti = 0; ti < 2; ++ti) {
    #pragma unroll
    for (int tj = 0; tj < 2; ++tj) {
      v8f cre = {};
      v8f cim = {};
      #pragma unroll
      for (int kk = 0; kk < 32; kk += 4) {
        const int ka = kk + 2 * hsel;
        const int arow = (ti * 16 + m0) * ST + ka;
        v2f are, aim, bre, bim;
        are.x = Are[arow];            are.y = Are[arow + 1];
        aim.x = Aim[arow];            aim.y = Aim[arow + 1];
        const int bcol = tj * 16 + m0;
        bre.x = Bre[ka * ST + bcol];  bre.y = Bre[(ka + 1) * ST + bcol];
        bim.x = Bim[ka * ST + bcol];  bim.y = Bim[(ka + 1) * ST + bcol];
        v2f naim; naim.x = -aim.x; naim.y = -aim.y;
        cre = wmma4(are,  bre, cre);   // +Are*Bre
        cre = wmma4(naim, bim, cre);   // -Aim*Bim
        cim = wmma4(are,  bim, cim);   // +Are*Bim
        cim = wmma4(aim,  bre, cim);   // +Aim*Bre
      }
      #pragma unroll
      for (int r = 0; r < 8; ++r) {
        const int row = ti * 16 + r + 8 * hsel;
        const int col = tj * 16 + m0;
        Dre[row * ST + col] = cre[r];
        Dim[row * ST + col] = cim[r];
      }
    }
  }
}

__global__ __launch_bounds__(32)
void qsim10_wmma_kernel(const float* __restrict__ x,
                        const float* __restrict__ theta,
                        const float* __restrict__ araw,
                        const float* __restrict__ braw,
                        const float* __restrict__ hw,
                        const float* __restrict__ hb,
                        const float* __restrict__ lscale,
                        float* __restrict__ out) {
  __shared__ float Mre[NH * ST], Mim[NH * ST];    // current 32x32 operator
  __shared__ float S0re[NH * ST], S0im[NH * ST];  // state ping
  __shared__ float S1re[NH * ST], S1im[NH * ST];  // state pong
  __shared__ float Gre[NQ][2][2], Gim[NQ][2][2];  // per-qubit 2x2 gate (RX*RY)

  const int lane = threadIdx.x;   // single-wave workgroup
  const int s    = blockIdx.x;    // sample id
  const float* xrow = x + s * DIM;
  __builtin_prefetch(xrow, 0, 1); // global_prefetch_b8

  // |00..0> : S[0][0] = 1
  for (int j = lane; j < NH * ST; j += 32) { S0re[j] = 0.f; S0im[j] = 0.f; }
  __syncthreads();
  if (lane == 0) S0re[0] = 1.f;
  __syncthreads();

  float* curRe = S0re; float* curIm = S0im;
  float* altRe = S1re; float* altIm = S1im;

  for (int b = 0; b < 3; ++b) {
    // ---- per-qubit combined gate G = RX(theta) * RY(enc angle) -------------
    if (lane < NQ) {
      const int q  = lane;
      const int fi = (b * NQ + q) % DIM;
      const float a = araw[fi];
      float alpha = (a > 20.f) ? a : log1pf(expf(a));   // softplus
      alpha += 1e-6f;
      const float beta = tanhf(braw[fi]);               // * ENC_BETA_MAX (=1)
      const float ang  = PI_F * (alpha * xrow[fi] + beta);
      const float hc = 0.5f * ang;
      const float c  = cosf(hc), sn = sinf(hc);
      const float ht = 0.5f * theta[b * NQ + q];
      const float ct = cosf(ht), st = sinf(ht);
      // RX*RY:
      Gre[q][0][0] =  ct * c;   Gim[q][0][0] = -st * sn;
      Gre[q][0][1] = -ct * sn;  Gim[q][0][1] = -st * c;
      Gre[q][1][0] =  ct * sn;  Gim[q][1][0] = -st * c;
      Gre[q][1][1] =  ct * c;   Gim[q][1][1] =  st * sn;
    }
    __syncthreads();

    // ---- build L = tensor product of G over high qubits 0..4 ---------------
    // L[h'][h] = prod_q G[q][bit_q(h')][bit_q(h)], qubit q -> bit (4-q) of h.
    for (int j = 0; j < 32; ++j) {
      const int e   = j * 32 + lane;
      const int row = e >> 5, col = e & 31;
      float pr = 1.f, pi2 = 0.f;
      #pragma unroll
      for (int qq = 0; qq < 5; ++qq) {
        const int i = (row >> (4 - qq)) & 1;
        const int k = (col >> (4 - qq)) & 1;
        const float gr = Gre[qq][i][k], gi = Gim[qq][i][k];
        const float nr = pr * gr - pi2 * gi;
        const float ni = pr * gi + pi2 * gr;
        pr = nr; pi2 = ni;
      }
      Mre[row * ST + col] = pr;
      Mim[row * ST + col] = pi2;
    }
    __syncthreads();
    cgemm32(Mre, Mim, curRe, curIm, altRe, altIm, lane);   // alt = L * S
    __syncthreads();

    // ---- build Rt (transposed right operator), low qubits 5..9 -------------
    // Rt[l][l'] = R[l'][l] = prod_q G[5+qq][bit(l')][bit(l)]
    for (int j = 0; j < 32; ++j) {
      const int e   = j * 32 + lane;
      const int row = e >> 5, col = e & 31;   // row = l (contraction), col = l'
      float pr = 1.f, pi2 = 0.f;
      #pragma unroll
      for (int qq = 0; qq < 5; ++qq) {
        const int i = (col >> (4 - qq)) & 1;  // output bit (from l')
        const int k = (row >> (4 - qq)) & 1;  // input bit (from l)
        const float gr = Gre[5 + qq][i][k], gi = Gim[5 + qq][i][k];
        const float nr = pr * gr - pi2 * gi;
        const float ni = pr * gi + pi2 * gr;
        pr = nr; pi2 = ni;
      }
      Mre[row * ST + col] = pr;
      Mim[row * ST + col] = pi2;
    }
    __syncthreads();
    cgemm32(altRe, altIm, Mre, Mim, curRe, curIm, lane);   // cur = (L*S) * Rt
    __syncthreads();

    // ---- CNOT chain: dest bits = prefix-XOR of source bits -----------------
    // Gather form: src_bit(q) = dst_bit(q) ^ dst_bit(q-1); qubit q = bit (9-q).
    for (int j = 0; j < 32; ++j) {
      const int m = j * 32 + lane;
      unsigned src = 0; int prevd = 0;
      #pragma unroll
      for (int q = 0; q < 10; ++q) {
        const int dq = (m >> (9 - q)) & 1;
        const int sq = dq ^ prevd;
        prevd = dq;
        src |= (unsigned)sq << (9 - q);
      }
      const int di = (m >> 5) * ST + (m & 31);
      const int si = (int)(src >> 5) * ST + (int)(src & 31);
      altRe[di] = curRe[si];
      altIm[di] = curIm[si];
    }
    __syncthreads();
    { float* t = curRe; curRe = altRe; altRe = t; }
    { float* t = curIm; curIm = altIm; altIm = t; }
  }

  // ---- measurement: <Z_q> = sum_n (+/-) |amp_n|^2 --------------------------
  float z[NQ];
  #pragma unroll
  for (int q = 0; q < NQ; ++q) z[q] = 0.f;
  for (int j = 0; j < 32; ++j) {
    const int n = j * 32 + lane;
    const int idx = (n >> 5) * ST + (n & 31);
    const float re = curRe[idx], im = curIm[idx];
    const float p = re * re + im * im;
    #pragma unroll
    for (int q = 0; q < NQ; ++q)
      z[q] += ((n >> (9 - q)) & 1) ? -p : p;
  }
  #pragma unroll
  for (int q = 0; q < NQ; ++q) {
    #pragma unroll
    for (int off = 16; off > 0; off >>= 1)
      z[q] += __shfl_xor(z[q], off, 32);
  }

  if (lane == 0) {
    float raw = hb[0];
    #pragma unroll
    for (int q = 0; q < NQ; ++q) raw += z[q] * hw[q];
    float sc = lscale[0];
    sc = fminf(fmaxf(sc, 0.5f), 80.f);
    float o = sc * raw;
    o = fminf(fmaxf(o, -30.f), 30.f);
    out[s] = o;
  }
}

extern "C" void kernel_launch(void* const* d_in, const int* in_sizes, int n_in,
                              void* d_out, int out_size, void* d_ws, size_t ws_size,
                              hipStream_t stream) {
  (void)in_sizes; (void)n_in; (void)d_ws; (void)ws_size;
  const float* x      = (const float*)d_in[0];
  const float* theta  = (const float*)d_in[1];
  const float* araw   = (const float*)d_in[2];
  const float* braw   = (const float*)d_in[3];
  const float* hw     = (const float*)d_in[4];
  const float* hb     = (const float*)d_in[5];
  const float* ls     = (const float*)d_in[6];
  float* out = (float*)d_out;
  // one single-wave workgroup per sample
  qsim10_wmma_kernel<<<dim3((unsigned)out_size), dim3(32), 0, stream>>>(
      x, theta, araw, braw, hw, hb, ls, out);
}